// W2Loss_77635828842952
// MI455X (gfx1250) — compile-verified
//
#include <hip/hip_runtime.h>
#include <cstdint>

// W2/transport-style loss on MI455X (gfx1250, wave32).
// Pipeline: k_wband (A^-1 band, analytic, padded) + k_zrhs (zero rhs halo)
// -> k_obs (obs row: renorm, scan, searchsorted -> q, rhs) -> k_solve (banded
// f32 WMMA GEMM, branch-free inner loop: M = A^-1 * rhs) -> k_loss (f row:
// renorm, scan, spline eval, trapz) -> k_final (fixed-order deterministic
// reduction). Rows of obs/f are staged into LDS with the TDM
// (tensor_load_to_lds + s_wait_tensorcnt). Needs ~53MB workspace.

#define BB   4096
#define NTT  2048
#define NPP  1024
#define NBLK 256

#define WB_COLS 96           // 16-col leading halo + 80-wide band
#define RHS_ROWS 1120        // 32 zero rows below + 1022 data + 66 zero above

typedef __attribute__((ext_vector_type(2))) float v2f;
typedef __attribute__((ext_vector_type(8))) float v8f;
typedef __attribute__((ext_vector_type(4))) unsigned int u32x4;
typedef __attribute__((ext_vector_type(8))) int i32x8;
typedef __attribute__((ext_vector_type(4))) int i32x4;

// ---------------- TDM: DMA one contiguous row of f32 into LDS ---------------
__device__ __forceinline__ void tdm_load_row_f32(void* lds_dst, const void* gsrc,
                                                 unsigned nelem) {
  unsigned lds_off = (unsigned)(uintptr_t)lds_dst;           // low 32b = LDS offset
  unsigned long long ga = (unsigned long long)(uintptr_t)gsrc;
  u32x4 g0;
  g0[0] = 1u;                                                // count=1, normal mode
  g0[1] = lds_off;                                           // lds_addr
  g0[2] = (unsigned)(ga & 0xFFFFFFFFull);                    // global_addr lo
  g0[3] = (unsigned)((ga >> 32) & 0x01FFFFFFull) | (2u << 30); // addr hi | type=2
  i32x8 g1;
  g1[0] = (int)(2u << 16);                                   // data_size = 4B
  g1[1] = (int)((nelem & 0xFFFFu) << 16);                    // tensor_dim0 lo16
  g1[2] = (int)(((nelem >> 16) & 0xFFFFu) | (1u << 16));     // dim0 hi16 | dim1=1
  g1[3] = (int)((nelem & 0xFFFFu) << 16);                    // tile_dim0
  g1[4] = 1;                                                 // tile_dim1=1, tile_dim2=0
  g1[5] = (int)nelem;                                        // tensor_dim0_stride lo
  g1[6] = 0;
  g1[7] = 0;
  i32x4 gz4 = {0, 0, 0, 0};
  i32x8 gz8 = {0, 0, 0, 0, 0, 0, 0, 0};
  // 6-arg toolchain variant: (g0, g1, g2, g3, extra, cpol)
  __builtin_amdgcn_tensor_load_to_lds(g0, g1, gz4, gz4, gz8, 0);
}

// ---------------- wave32 block helpers (256 threads = 8 waves) --------------
__device__ __forceinline__ float blockReduceSum(float v, float* scr) {
  const int lane = threadIdx.x & 31;
#pragma unroll
  for (int o = 16; o > 0; o >>= 1) v += __shfl_down(v, o, 32);
  if (lane == 0) scr[threadIdx.x >> 5] = v;
  __syncthreads();
  if (threadIdx.x == 0) {
    float s = 0.f;
    for (int w = 0; w < 8; ++w) s += scr[w];   // fixed order: deterministic
    scr[32] = s;
  }
  __syncthreads();
  float r = scr[32];
  __syncthreads();
  return r;
}

// Inclusive scan of 8 elems/thread over 2048; result back in v[], optionally
// mirrored to LDS array.
__device__ __forceinline__ void blockScan8(float (&v)[8], float* outArr, int base,
                                           float* scr) {
  float run = 0.f, loc[8];
#pragma unroll
  for (int e = 0; e < 8; ++e) { run += v[e]; loc[e] = run; }
  const int lane = threadIdx.x & 31, wid = threadIdx.x >> 5;
  float ts = run;
#pragma unroll
  for (int o = 1; o < 32; o <<= 1) {
    float n = __shfl_up(ts, o, 32);
    if (lane >= o) ts += n;
  }
  if (lane == 31) scr[wid] = ts;
  __syncthreads();
  if (threadIdx.x == 0) {
    float s = 0.f;
    for (int w = 0; w < 8; ++w) { float x = scr[w]; scr[8 + w] = s; s += x; }
  }
  __syncthreads();
  const float off = scr[8 + wid] + (ts - run);
#pragma unroll
  for (int e = 0; e < 8; ++e) {
    v[e] = off + loc[e];
    if (outArr) outArr[base + e] = v[e];
  }
  __syncthreads();
}

// ---------------- kernel: padded band of A^-1 for tridiag(1,4,1) ------------
// wbPad[i][c] = Ainv[i][i + c - 48] for i<1022 and valid j, else 0.
// Analytic stable form, theta = ln(2+sqrt(3)).
__global__ void k_wband(float* __restrict__ wb) {
  const int idx = blockIdx.x * blockDim.x + threadIdx.x;
  const int M = NPP - 2;
  if (idx >= NPP * WB_COLS) return;
  const int i = idx / WB_COLS, c = idx % WB_COLS;
  const int j = i + c - 48;
  float val = 0.f;
  if (i < M && j >= 0 && j < M) {
    const float th = 1.3169578969248166f;
    const int r = (i < j ? i : j) + 1;          // 1-indexed min
    const int s = (i > j ? i : j) + 1;          // 1-indexed max
    float num = expf(-th * (float)(s - r + 1)) *
                (1.f - expf(-2.f * th * (float)r)) *
                (1.f - expf(-2.f * th * (float)(M + 1 - s)));
    val = num / 0.9282032302755088f;            // 1 - e^{-2 theta}
    if ((i + j) & 1) val = -val;
  }
  wb[idx] = val;
}

// ---------------- kernel: zero the rhsPad halo rows -------------------------
// Rows [0,32) and [1054,1120) of rhsPad must read as 0 every launch.
__global__ void k_zrhs(float* __restrict__ rhsP) {
  const int idx = blockIdx.x * blockDim.x + threadIdx.x;
  const int NHALO = (32 + (RHS_ROWS - (32 + NPP - 2))) * BB;  // 98 rows
  if (idx >= NHALO) return;
  const int row = idx / BB, col = idx % BB;
  const int rp = (row < 32) ? row : (1054 + (row - 32));
  rhsP[(size_t)rp * BB + col] = 0.f;
}

// ---------------- kernel: obs row -> q row + rhsPad -------------------------
__global__ __launch_bounds__(NBLK) void k_obs(const float* __restrict__ obs,
                                              const float* __restrict__ t,
                                              const float* __restrict__ p,
                                              float* __restrict__ q,
                                              float* __restrict__ rhsP) {
  __shared__ __align__(16) float sA[NTT];
  __shared__ __align__(16) float cdf[NTT];
  __shared__ float qS[NPP];
  __shared__ float scr[64];
  const int b = blockIdx.x, tid = threadIdx.x;
  const float dt = t[1] - t[0];

  if ((tid >> 5) == 0) {                        // wave 0 issues the DMA
    tdm_load_row_f32(sA, obs + (size_t)b * NTT, NTT);
    __builtin_amdgcn_s_wait_tensorcnt(0);
  }
  __syncthreads();

  const float4 r0 = *reinterpret_cast<const float4*>(&sA[tid * 8]);
  const float4 r1 = *reinterpret_cast<const float4*>(&sA[tid * 8 + 4]);
  float sq[8] = {r0.x * r0.x, r0.y * r0.y, r0.z * r0.z, r0.w * r0.w,
                 r1.x * r1.x, r1.y * r1.y, r1.z * r1.z, r1.w * r1.w};
  float part = 0.f;
#pragma unroll
  for (int e = 0; e < 8; ++e) { part += sq[e]; sA[tid * 8 + e] = sq[e]; }
  const float tot = blockReduceSum(part, scr);
  const float Z = dt * (tot - 0.5f * (sA[0] + sA[NTT - 1]));   // trapz(sq)

  float tl[8];
#pragma unroll
  for (int e = 0; e < 8; ++e) { tl[e] = sq[e] / Z; sA[tid * 8 + e] = tl[e]; }
  __syncthreads();

  float inc[8];
#pragma unroll
  for (int e = 0; e < 8; ++e) {
    const int j = tid * 8 + e;
    inc[e] = (j == 0) ? 0.f : (sA[j - 1] + tl[e]) * (0.5f * dt);
  }
  blockScan8(inc, cdf, tid * 8, scr);           // cdf_obs in LDS

  // searchsorted(cdf, p, 'left'), clip, q = t[idx]
#pragma unroll
  for (int r = 0; r < 4; ++r) {
    const int k = tid + NBLK * r;
    const float v = p[k];
    int lo = 0, hi = NTT;
    while (lo < hi) {
      const int mid = (lo + hi) >> 1;
      if (cdf[mid] < v) lo = mid + 1; else hi = mid;
    }
    const int idx = lo < NTT ? lo : NTT - 1;
    const float qk = t[idx];
    qS[k] = qk;
    q[(size_t)b * NPP + k] = qk;
  }
  __syncthreads();

  const float h = p[1] - p[0];
#pragma unroll
  for (int r = 0; r < 4; ++r) {
    const int k = tid + NBLK * r;
    if (k < NPP - 2)
      rhsP[(size_t)(k + 32) * BB + b] =
          6.0f * ((qS[k] - 2.0f * qS[k + 1]) + qS[k + 2]) / (h * h);
  }
}

// ---------------- kernel: banded f32 WMMA GEMM  M = A^-1 * rhs --------------
// Per block: 16-batch tile; wave w handles i-blocks w, w+8, ... (64 total).
// Padded operands => branch-free inner loop (loads always in-bounds).
__global__ __launch_bounds__(NBLK) void k_solve(const float* __restrict__ rhsP,
                                                const float* __restrict__ wb,
                                                float* __restrict__ Mbuf) {
  const int nb0 = blockIdx.x * 16;              // batch tile base
  const int lane = threadIdx.x & 31, w = threadIdx.x >> 5;
  const int Mr = lane & 15, ks = lane >> 4;     // A: rows / K-pair select
  const int Mint = NPP - 2;                     // 1022
  const int nb = nb0 + (lane & 15);             // B/D column (batch)

  for (int ib = w; ib < 64; ib += 8) {
    const int i0 = ib * 16;
    const int iA = i0 + Mr;                     // A row (padded table: 0..1023)
    v8f acc = {0.f, 0.f, 0.f, 0.f, 0.f, 0.f, 0.f, 0.f};
#pragma unroll
    for (int kk = 0; kk < 20; ++kk) {
      const int jb = i0 - 32 + kk * 4 + 2 * ks; // this lane's K indices jb, jb+1
      const int wc = jb - iA + 48;              // in [1,95] by construction
      const float a0 = wb[iA * WB_COLS + wc];
      const float a1 = wb[iA * WB_COLS + wc + 1];
      const size_t ro = (size_t)(jb + 32) * BB + nb;  // rp in [0,1087]
      const float b0 = rhsP[ro];
      const float b1 = rhsP[ro + BB];
      v2f A = {a0, a1};
      v2f Bm = {b0, b1};
      acc = __builtin_amdgcn_wmma_f32_16x16x4_f32(false, A, false, Bm,
                                                  (short)0, acc, false, false);
    }
    // D layout: VGPR v -> rows v (lanes 0-15) / v+8 (lanes 16-31), col = lane&15.
#pragma unroll
    for (int v = 0; v < 8; ++v) {
      const int irow = i0 + v + 8 * ks;
      if (irow < Mint)
        Mbuf[(size_t)nb * NPP + irow + 1] = acc[v];
    }
  }
}

// ---------------- kernel: f row -> per-row loss -----------------------------
__global__ __launch_bounds__(NBLK) void k_loss(const float* __restrict__ f,
                                               const float* __restrict__ t,
                                               const float* __restrict__ p,
                                               const float* __restrict__ q,
                                               const float* __restrict__ Mbuf,
                                               float* __restrict__ rowloss) {
  __shared__ __align__(16) float fS[NTT];
  __shared__ float FS[NTT];
  __shared__ float pS[NPP];
  __shared__ float qS[NPP];
  __shared__ float MS[NPP];
  __shared__ float scr[64];
  const int b = blockIdx.x, tid = threadIdx.x;
  const float dt = t[1] - t[0];
  const float h = p[1] - p[0];

  if ((tid >> 5) == 0) {
    tdm_load_row_f32(fS, f + (size_t)b * NTT, NTT);
    __builtin_amdgcn_s_wait_tensorcnt(0);
  }
#pragma unroll
  for (int r = 0; r < 4; ++r) {                 // meanwhile: spline tables
    const int k = tid + NBLK * r;
    pS[k] = p[k];
    qS[k] = q[(size_t)b * NPP + k];
    MS[k] = (k == 0 || k == NPP - 1) ? 0.f : Mbuf[(size_t)b * NPP + k];
  }
  __syncthreads();

  const float4 r0 = *reinterpret_cast<const float4*>(&fS[tid * 8]);
  const float4 r1 = *reinterpret_cast<const float4*>(&fS[tid * 8 + 4]);
  float sq[8] = {r0.x * r0.x, r0.y * r0.y, r0.z * r0.z, r0.w * r0.w,
                 r1.x * r1.x, r1.y * r1.y, r1.z * r1.z, r1.w * r1.w};
  float part = 0.f;
#pragma unroll
  for (int e = 0; e < 8; ++e) { part += sq[e]; fS[tid * 8 + e] = sq[e]; }
  const float tot = blockReduceSum(part, scr);
  const float Zf = dt * (tot - 0.5f * (fS[0] + fS[NTT - 1]));

  float tl[8];
#pragma unroll
  for (int e = 0; e < 8; ++e) { tl[e] = sq[e] / Zf; fS[tid * 8 + e] = tl[e]; }
  __syncthreads();

  float inc[8];
#pragma unroll
  for (int e = 0; e < 8; ++e) {
    const int j = tid * 8 + e;
    inc[e] = (j == 0) ? 0.f : (fS[j - 1] + tl[e]) * (0.5f * dt);
  }
  blockScan8(inc, FS, tid * 8, scr);            // F in inc[] now

  float contrib = 0.f;
#pragma unroll
  for (int e = 0; e < 8; ++e) {
    const int j = tid * 8 + e;
    const float x = inc[e];                     // F_j
    int lo = 0, hi = NPP;                       // searchsorted(p, x, 'right')
    while (lo < hi) {
      const int mid = (lo + hi) >> 1;
      if (pS[mid] <= x) lo = mid + 1; else hi = mid;
    }
    int k = lo - 1;
    k = k < 0 ? 0 : (k > NPP - 2 ? NPP - 2 : k);
    const float s = x - pS[k];
    const float ag = qS[k];
    const float bg = (qS[k + 1] - qS[k]) / h - h * (2.0f * MS[k] + MS[k + 1]) / 6.0f;
    const float cg = MS[k] * 0.5f;
    const float dg = (MS[k + 1] - MS[k]) / (6.0f * h);
    const float od = ag + s * (bg + s * (cg + s * dg));
    const float tj = (float)j * dt;
    const float dif = tj - od;
    const float g = dif * dif * tl[e];
    contrib += g * ((j == 0 || j == NTT - 1) ? 0.5f : 1.0f);
  }
  const float rsum = blockReduceSum(contrib, scr);
  if (tid == 0) rowloss[b] = dt * rsum;
}

// ---------------- kernel: deterministic final reduction ---------------------
__global__ __launch_bounds__(NBLK) void k_final(const float* __restrict__ rowloss,
                                                float* __restrict__ out) {
  __shared__ float scr[64];
  float s = 0.f;
  for (int i = threadIdx.x; i < BB; i += NBLK) s += rowloss[i];
  const float tot = blockReduceSum(s, scr);
  if (threadIdx.x == 0) out[0] = tot;
}

extern "C" void kernel_launch(void* const* d_in, const int* in_sizes, int n_in,
                              void* d_out, int out_size, void* d_ws, size_t ws_size,
                              hipStream_t stream) {
  (void)in_sizes; (void)n_in; (void)out_size; (void)ws_size;
  const float* f   = (const float*)d_in[0];
  const float* obs = (const float*)d_in[1];
  const float* t   = (const float*)d_in[2];
  const float* p   = (const float*)d_in[3];

  float* ws   = (float*)d_ws;                        // needs ~53 MB
  float* q    = ws;                                  // BB*NPP
  float* rhsP = q    + (size_t)BB * NPP;             // RHS_ROWS*BB (padded)
  float* Mbuf = rhsP + (size_t)RHS_ROWS * BB;        // BB*NPP (cols 1..1022 written)
  float* wb   = Mbuf + (size_t)BB * NPP;             // NPP*WB_COLS (padded)
  float* rls  = wb   + (size_t)NPP * WB_COLS;        // BB

  k_wband<<<(NPP * WB_COLS + NBLK - 1) / NBLK, NBLK, 0, stream>>>(wb);
  k_zrhs <<<(98 * BB + NBLK - 1) / NBLK, NBLK, 0, stream>>>(rhsP);
  k_obs  <<<BB, NBLK, 0, stream>>>(obs, t, p, q, rhsP);
  k_solve<<<BB / 16, NBLK, 0, stream>>>(rhsP, wb, Mbuf);
  k_loss <<<BB, NBLK, 0, stream>>>(f, t, p, q, Mbuf, rls);
  k_final<<<1, NBLK, 0, stream>>>(rls, (float*)d_out);
}